// SOM_30425548325146
// MI455X (gfx1250) — compile-verified
//
#include <hip/hip_runtime.h>
#include <stdint.h>

typedef __attribute__((ext_vector_type(2))) float v2f;
typedef __attribute__((ext_vector_type(8))) float v8f;

#define X_SIZE 512
#define Y_SIZE 512
#define W_SIZE 256
#define N_CELLS (X_SIZE * Y_SIZE)          // 262144
#define N_STRIPS (N_CELLS / 16)            // 16384 strips of 16 cells
#define BLOCKS 512
#define WAVES_PER_BLOCK 8
#define STRIPS_PER_WAVE (N_STRIPS / (BLOCKS * WAVES_PER_BLOCK))  // 4

__device__ __forceinline__ unsigned orderedKey(float f) {
    // monotone mapping of IEEE float to unsigned (for atomicMin)
    unsigned u = __float_as_uint(f);
    return (u & 0x80000000u) ? ~u : (u | 0x80000000u);
}

__global__ void som_init(unsigned long long* ws) {
    *ws = 0xFFFFFFFFFFFFFFFFull;
}

// Pass 1: streaming pass over W. Per 16-cell strip:
//   dot(x, W_cell)   -> V_WMMA_F32_16X16X4_F32 accumulation (x broadcast over columns)
//   ||W_cell||^2     -> co-executing VALU FMAs on the same A registers
//   score = ||W||^2 - 2*dot  (equals d^2 - ||x||^2, same argmin)
__global__ __launch_bounds__(256) void som_argmin(const float* __restrict__ x,
                                                  const float* __restrict__ W,
                                                  unsigned long long* __restrict__ ws) {
    __shared__ float xs[W_SIZE];
    __shared__ float dots[WAVES_PER_BLOCK][16];

    const int tid = threadIdx.x;
    xs[tid] = x[tid];              // stage x in LDS (256 threads, 256 floats)
    __syncthreads();

    const int lane = tid & 31;
    const int wave = tid >> 5;
    const int gwave = blockIdx.x * WAVES_PER_BLOCK + wave;
    const int halfOff = (lane < 16) ? 0 : 2;   // K sub-offset per ISA 16x4 f32 A layout
    const int m = lane & 15;                   // cell row within strip

    float bestScore = 3.0e38f;
    int bestIdx = 0;

    for (int s = 0; s < STRIPS_PER_WAVE; ++s) {
        const int strip = gwave * STRIPS_PER_WAVE + s;
        const float* wBase = W + (size_t)strip * 16u * W_SIZE
                               + (size_t)m * W_SIZE + (size_t)halfOff;

        v8f acc = {};
        float sumsq = 0.0f;

#pragma unroll 8
        for (int c = 0; c < W_SIZE / 4; ++c) {
            // A: 16 cells x 4 K of W (per-lane b64 load, K pair per lane)
            v2f a = *(const v2f*)(wBase + 4 * c);
            // B: 4 x 16 with x chunk replicated across columns (same lane->K map)
            v2f b;
            b.x = xs[4 * c + halfOff];
            b.y = xs[4 * c + halfOff + 1];
            acc = __builtin_amdgcn_wmma_f32_16x16x4_f32(
                false, a, false, b, (short)0, acc, false, false);
            // sum of squares on the same loaded values (per half-wave partial)
            sumsq = fmaf(a.x, a.x, sumsq);
            sumsq = fmaf(a.y, a.y, sumsq);
        }

        // combine the two half-wave K partials of ||W||^2
        sumsq += __shfl_xor(sumsq, 16, 32);

        // Dump one column of the accumulator into LDS.
        // C layout: VGPR r holds M=r (lanes 0-15) and M=8+r (lanes 16-31);
        // all 16 columns are identical, so lanes 0 and 16 each export 8 cells.
        if (m == 0) {
            const int base = (lane >> 4) * 8;
#pragma unroll
            for (int r = 0; r < 8; ++r) dots[wave][base + r] = acc[r];
        }
        // same-wave DS ops are in-order: store above is visible to the load below
        const float score = fmaf(-2.0f, dots[wave][m], sumsq);
        const int idx = strip * 16 + m;
        if (lane < 16) {
            if (score < bestScore) { bestScore = score; bestIdx = idx; }
        }
    }

    // wave min-reduce (lanes 16..31 carry +inf sentinels, harmless)
#pragma unroll
    for (int off = 16; off >= 1; off >>= 1) {
        float os = __shfl_xor(bestScore, off, 32);
        int   oi = __shfl_xor(bestIdx, off, 32);
        if (os < bestScore || (os == bestScore && oi < bestIdx)) {
            bestScore = os; bestIdx = oi;
        }
    }
    if (lane == 0) {
        unsigned long long key =
            ((unsigned long long)orderedKey(bestScore) << 32) | (unsigned)bestIdx;
        atomicMin(ws, key);
    }
}

// Pass 2: Gaussian neighborhood. Reference's transposes collapse to
// h[i][j] = exp(-((i-wx)^2 + (j-wy)^2) / (2*decay^2)).
__global__ __launch_bounds__(256) void som_h(const unsigned long long* __restrict__ ws,
                                             const int* __restrict__ time_step,
                                             float* __restrict__ out, int out_size) {
    const unsigned idx = (unsigned)(*ws & 0xFFFFFFFFull);
    const int wx = (int)(idx >> 9);
    const int wy = (int)(idx & 511u);

    const float SIGMA = 2.0f;
    const float TIME_CONST = 1000.0f / logf(SIGMA);
    const float t = (float)(*time_step);
    const float decay = SIGMA * expf(-t / TIME_CONST);
    const float denom = 2.0f * decay * decay;

    const int gid = blockIdx.x * blockDim.x + threadIdx.x;
    const int i = gid >> 9;
    const int j = gid & 511;
    const float dx = (float)(i - wx);
    const float dy = (float)(j - wy);
    out[gid] = expf(-(dx * dx + dy * dy) / denom);

    if (gid == 0 && out_size >= N_CELLS + 2) {
        out[N_CELLS]     = (float)wx;
        out[N_CELLS + 1] = (float)wy;
    }
}

extern "C" void kernel_launch(void* const* d_in, const int* in_sizes, int n_in,
                              void* d_out, int out_size, void* d_ws, size_t ws_size,
                              hipStream_t stream) {
    const float* x = (const float*)d_in[0];
    const float* W = (const float*)d_in[1];
    const int* ts  = (const int*)d_in[2];
    unsigned long long* ws = (unsigned long long*)d_ws;

    som_init<<<1, 1, 0, stream>>>(ws);
    som_argmin<<<BLOCKS, 256, 0, stream>>>(x, W, ws);
    som_h<<<N_CELLS / 256, 256, 0, stream>>>(ws, ts, (float*)d_out, out_size);
}